// UpdateEdgeBlock_6734508720701
// MI455X (gfx1250) — compile-verified
//
#include <hip/hip_runtime.h>
#include <math.h>
#include <stdint.h>

typedef __attribute__((ext_vector_type(2))) float v2f;
typedef __attribute__((ext_vector_type(8))) float v8f;

#define C_DIM     32
#define NBASIS    8
#define RCUT      5.0f
#define EPSV      1e-8f
#define NCOMBO    11
#define TILE_E    16
#define LDS_STR   33      // padded channel stride (avoid 32-float bank conflicts)
#define NCOMP     13      // 1 (l=0) + 3 (l=1) + 9 (l=2)
#define PI_F      3.14159265358979323846f

__device__ __forceinline__ float silu_f(float x) {
    return x * (1.0f / (1.0f + __expf(-x)));
}

__global__ __launch_bounds__(256)
void edge_update_kernel(
    const float* __restrict__ node0, const float* __restrict__ node1,
    const float* __restrict__ node2,
    const float* __restrict__ edge0, const float* __restrict__ edge1,
    const float* __restrict__ edge2,
    const float* __restrict__ rij,   const float* __restrict__ W_rad,
    const float* __restrict__ Wsi0,  const float* __restrict__ Wsi1,
    const float* __restrict__ Wsi2,  const float* __restrict__ bsi0,
    const float* __restrict__ nlw1,  const float* __restrict__ nlb1,
    const float* __restrict__ nlw2,  const float* __restrict__ nlb2,
    const int*   __restrict__ idx_j,
    float* __restrict__ out0, float* __restrict__ out1, float* __restrict__ out2,
    int E)
{
    __shared__ float msgL[NCOMP * TILE_E * LDS_STR];
    __shared__ float sL  [NCOMP * TILE_E * LDS_STR];
    __shared__ __attribute__((aligned(16))) float edge2L[TILE_E * C_DIM * 9];

    const int tid  = threadIdx.x;
    const int lane = tid & 31;
    const int wave = tid >> 5;
    const int e0   = blockIdx.x * TILE_E;

    // ----------------- Phase 0: async-stage edge2 residual tile (18KB) into
    // LDS via ASYNCcnt-tracked global_load_async_to_lds_b128; overlaps with
    // phases 1-2. Also prefetch the smaller edge0/edge1 residual streams.
    {
        const int ne     = (E - e0) < TILE_E ? (E - e0) : TILE_E;
        const int nbytes = ne * C_DIM * 9 * 4;
        const uint32_t lds_base = (uint32_t)(size_t)(&edge2L[0]);
        const uint64_t g_base   = (uint64_t)(size_t)(edge2 + (size_t)e0 * (C_DIM * 9));
        for (int off = tid * 16; off < nbytes; off += 256 * 16) {
            uint32_t l = lds_base + (uint32_t)off;
            uint64_t g = g_base + (uint64_t)off;
            asm volatile("global_load_async_to_lds_b128 %0, %1, off"
                         :: "v"(l), "v"(g) : "memory");
        }
        if (tid < 16)
            __builtin_prefetch((const char*)(edge0 + (size_t)e0 * C_DIM) + tid * 128, 0, 3);
        if (tid < 48)
            __builtin_prefetch((const char*)(edge1 + (size_t)e0 * C_DIM * 3) + tid * 128, 0, 3);
    }

    // ----------------- Phase 1: build equivariant messages (lane == channel)
    #pragma unroll
    for (int rep = 0; rep < 2; ++rep) {
        const int m = wave + rep * 8;              // edge slot within tile
        int e = e0 + m;
        if (e >= E) e = E - 1;                     // clamp reads (writes to LDS harmless)
        const int c = lane;

        const float rx = rij[e * 3 + 0];
        const float ry = rij[e * 3 + 1];
        const float rz = rij[e * 3 + 2];
        const float d     = sqrtf(rx * rx + ry * ry + rz * rz + EPSV);
        const float inv_d = 1.0f / d;
        float u[3] = { rx * inv_d, ry * inv_d, rz * inv_d };

        const float dc = fminf(d, RCUT);
        const float fc = 0.5f * (__cosf(PI_F * dc / RCUT) + 1.0f);
        float rbf[NBASIS];
        #pragma unroll
        for (int b = 0; b < NBASIS; ++b)
            rbf[b] = __sinf((float)(b + 1) * PI_F * d / RCUT) * inv_d * fc;

        const int j = idx_j[e];
        const float g0 = node0[j * C_DIM + c];
        float g1[3];
        #pragma unroll
        for (int i = 0; i < 3; ++i) g1[i] = node1[(j * C_DIM + c) * 3 + i];
        float g2[9];
        #pragma unroll
        for (int i = 0; i < 9; ++i) g2[i] = node2[(j * C_DIM + c) * 9 + i];

        float filt[NCOMBO];
        #pragma unroll
        for (int ci = 0; ci < NCOMBO; ++ci) {
            float acc = 0.0f;
            #pragma unroll
            for (int b = 0; b < NBASIS; ++b)
                acc += rbf[b] * W_rad[b * (NCOMBO * C_DIM) + ci * C_DIM + c];
            filt[ci] = acc;
        }

        float msg0 = 0.0f;
        float msg1[3] = {0.f, 0.f, 0.f};
        float msg2[9] = {0.f, 0.f, 0.f, 0.f, 0.f, 0.f, 0.f, 0.f, 0.f};

        // (0,0,0)
        msg0 += g0 * filt[0];
        // (0,1,1)
        #pragma unroll
        for (int i = 0; i < 3; ++i) msg1[i] += g0 * filt[1] * u[i];
        // (0,2,2)
        #pragma unroll
        for (int i = 0; i < 3; ++i)
            #pragma unroll
            for (int k = 0; k < 3; ++k) msg2[3 * i + k] += g0 * filt[2] * u[i] * u[k];
        // (1,0,1)
        #pragma unroll
        for (int i = 0; i < 3; ++i) msg1[i] += g1[i] * filt[3];
        // (1,1,0) + (1,2,1) share g1.u
        {
            float dot = g1[0] * u[0] + g1[1] * u[1] + g1[2] * u[2];
            msg0 += filt[4] * dot;
            #pragma unroll
            for (int k = 0; k < 3; ++k) msg1[k] += filt[6] * dot * u[k];
        }
        // (1,1,2)
        #pragma unroll
        for (int i = 0; i < 3; ++i)
            #pragma unroll
            for (int k = 0; k < 3; ++k) msg2[3 * i + k] += g1[i] * filt[5] * u[k];
        // (2,0,2)
        #pragma unroll
        for (int q = 0; q < 9; ++q) msg2[q] += g2[q] * filt[7];
        // (2,1,1), (2,2,0), (2,2,2): all use w[i] = sum_j g2[i][j] u[j]
        {
            float w[3];
            #pragma unroll
            for (int i = 0; i < 3; ++i)
                w[i] = g2[3 * i + 0] * u[0] + g2[3 * i + 1] * u[1] + g2[3 * i + 2] * u[2];
            #pragma unroll
            for (int i = 0; i < 3; ++i) msg1[i] += filt[8] * w[i];
            float quad = w[0] * u[0] + w[1] * u[1] + w[2] * u[2];
            msg0 += filt[9] * quad;
            #pragma unroll
            for (int i = 0; i < 3; ++i)
                #pragma unroll
                for (int k = 0; k < 3; ++k) msg2[3 * i + k] += filt[10] * w[i] * u[k];
        }

        msgL[(0 * TILE_E + m) * LDS_STR + c] = msg0;
        #pragma unroll
        for (int i = 0; i < 3; ++i) msgL[((1 + i) * TILE_E + m) * LDS_STR + c] = msg1[i];
        #pragma unroll
        for (int q = 0; q < 9; ++q) msgL[((4 + q) * TILE_E + m) * LDS_STR + c] = msg2[q];
    }

    __syncthreads();

    // ----------------- Phase 2: channel-mix GEMMs via V_WMMA_F32_16X16X4_F32
    // 26 tiles = 13 comps x 2 N-tiles; trip count uniform per wave -> EXEC all 1s.
    const int mrow  = lane & 15;
    const int khalf = (lane >> 4) * 2;             // lanes 16-31 take K+2, K+3
    for (int t = wave; t < 2 * NCOMP; t += 8) {
        const int comp  = t >> 1;
        const int ntile = t & 1;
        const float* __restrict__ W = (comp == 0) ? Wsi0 : ((comp < 4) ? Wsi1 : Wsi2);
        const int ncol = ntile * 16 + (lane & 15);
        const float* msgc = &msgL[comp * TILE_E * LDS_STR];

        v8f acc = {};
        #pragma unroll
        for (int kk = 0; kk < 8; ++kk) {
            const int kb = kk * 4 + khalf;
            v2f a, b;
            a.x = msgc[mrow * LDS_STR + kb];           // A: 16 edges x 4 channels
            a.y = msgc[mrow * LDS_STR + kb + 1];
            b.x = W[(kb)     * C_DIM + ncol];          // B: 4 channels x 16 outs
            b.y = W[(kb + 1) * C_DIM + ncol];
            acc = __builtin_amdgcn_wmma_f32_16x16x4_f32(
                false, a, false, b, (short)0, acc, false, false);
        }

        float* sc = &sL[comp * TILE_E * LDS_STR];
        #pragma unroll
        for (int v = 0; v < 8; ++v) {
            const int mo = v + 8 * (lane >> 4);        // D layout: VGPR v -> rows v / v+8
            sc[mo * LDS_STR + ntile * 16 + (lane & 15)] = acc[v];
        }
    }

    // Drain this wave's async edge2 copies, then make all waves' LDS writes
    // (s tiles + staged edge2) visible.
    asm volatile("s_wait_asynccnt 0x0" ::: "memory");
    __syncthreads();

    // ----------------- Phase 3: gated nonlinearities + residual add
    {
        const int dch = tid & 31;
        #pragma unroll
        for (int rep = 0; rep < 2; ++rep) {
            const int m = (tid >> 5) + rep * 8;
            const int e = e0 + m;
            if (e >= E) continue;

            const float s0 = sL[(0 * TILE_E + m) * LDS_STR + dch] + bsi0[dch];
            out0[e * C_DIM + dch] = edge0[e * C_DIM + dch] + silu_f(s0);

            float s1v[3];
            float acc1 = EPSV;
            #pragma unroll
            for (int i = 0; i < 3; ++i) {
                s1v[i] = sL[((1 + i) * TILE_E + m) * LDS_STR + dch];
                acc1 += s1v[i] * s1v[i];
            }
            const float n1 = sqrtf(acc1);
            const float gate1 = silu_f(nlw1[dch] * n1 + nlb1[dch]);
            #pragma unroll
            for (int i = 0; i < 3; ++i)
                out1[(e * C_DIM + dch) * 3 + i] =
                    edge1[(e * C_DIM + dch) * 3 + i] + gate1 * s1v[i];

            float s2v[9];
            float acc2 = EPSV;
            #pragma unroll
            for (int q = 0; q < 9; ++q) {
                s2v[q] = sL[((4 + q) * TILE_E + m) * LDS_STR + dch];
                acc2 += s2v[q] * s2v[q];
            }
            const float n2 = sqrtf(acc2);
            const float gate2 = silu_f(nlw2[dch] * n2 + nlb2[dch]);
            #pragma unroll
            for (int q = 0; q < 9; ++q)
                out2[(e * C_DIM + dch) * 9 + q] =
                    edge2L[(m * C_DIM + dch) * 9 + q] + gate2 * s2v[q];
        }
    }
}

extern "C" void kernel_launch(void* const* d_in, const int* in_sizes, int n_in,
                              void* d_out, int out_size, void* d_ws, size_t ws_size,
                              hipStream_t stream) {
    const float* node0 = (const float*)d_in[0];
    const float* node1 = (const float*)d_in[1];
    const float* node2 = (const float*)d_in[2];
    const float* edge0 = (const float*)d_in[3];
    const float* edge1 = (const float*)d_in[4];
    const float* edge2 = (const float*)d_in[5];
    const float* rij   = (const float*)d_in[6];
    const float* W_rad = (const float*)d_in[7];
    const float* Wsi0  = (const float*)d_in[8];
    const float* Wsi1  = (const float*)d_in[9];
    const float* Wsi2  = (const float*)d_in[10];
    const float* bsi0  = (const float*)d_in[11];
    const float* nlw1  = (const float*)d_in[12];
    const float* nlb1  = (const float*)d_in[13];
    const float* nlw2  = (const float*)d_in[14];
    const float* nlb2  = (const float*)d_in[15];
    const int*   idxj  = (const int*)d_in[16];

    const int E = in_sizes[16];
    float* out0 = (float*)d_out;
    float* out1 = out0 + (size_t)E * C_DIM;
    float* out2 = out1 + (size_t)E * C_DIM * 3;

    const int nblocks = (E + TILE_E - 1) / TILE_E;
    edge_update_kernel<<<nblocks, 256, 0, stream>>>(
        node0, node1, node2, edge0, edge1, edge2, rij, W_rad,
        Wsi0, Wsi1, Wsi2, bsi0, nlw1, nlb1, nlw2, nlb2, idxj,
        out0, out1, out2, E);
}